// WaveletActivation_60705067761825
// MI455X (gfx1250) — compile-verified
//
#include <hip/hip_runtime.h>

// Haar 12-level cascade on a 4096-wide last axis collapses to rowsum/64.
// Bandwidth-bound (0.25 FLOP/byte): stream with coalesced non-temporal b128
// loads, one wave32 per row, and do the 32-lane partial reduction with two
// V_WMMA_F32_16X16X4_F32 ones-matrix folds (exact f32 math, EXEC all-1s).

typedef float v2f __attribute__((ext_vector_type(2)));
typedef float v4f __attribute__((ext_vector_type(4)));
typedef float v8f __attribute__((ext_vector_type(8)));

// Reduce the 32 per-lane partials of this wave to a single value, returned
// uniformly in all lanes. Uses D = A x B with B = ones(4x16):
//   fold1: D[m][n] = p[m] + p[m+16]        (32 -> 16, in D-VGPRs)
//   valu : t = sum of 8 D-VGPRs            (16 -> 2 half-wave sums)
//   fold2: D[m][n] = t_lo + t_hi = total   (2 -> 1)
__device__ __forceinline__ float wave_reduce_f32_wmma(float p) {
  v2f ones; ones.x = 1.0f; ones.y = 1.0f;
  v8f c = {};

  v2f a; a.x = p; a.y = 0.0f;
  v8f d = __builtin_amdgcn_wmma_f32_16x16x4_f32(
      /*neg_a=*/false, a, /*neg_b=*/false, ones,
      /*c_mod=*/(short)0, c, /*reuse_a=*/false, /*reuse_b=*/false);

  float t = ((d[0] + d[1]) + (d[2] + d[3])) + ((d[4] + d[5]) + (d[6] + d[7]));

  v2f a2; a2.x = t; a2.y = 0.0f;
  v8f d2 = __builtin_amdgcn_wmma_f32_16x16x4_f32(
      false, a2, false, ones, (short)0, c, false, false);
  return d2[0];
}

__global__ __launch_bounds__(256, 1)
void WaveletActivation_rowsum_kernel(const float* __restrict__ x,
                                     float* __restrict__ out,
                                     int nrows, int cols) {
  const int wave = threadIdx.x >> 5;
  const int lane = threadIdx.x & 31;

  long long row  = (long long)blockIdx.x * 8 + wave;
  // Clamp for loads so EXEC stays all-ones through the WMMAs (store is guarded).
  long long rowc = row < nrows ? row : (long long)nrows - 1;

  const v4f* __restrict__ src =
      (const v4f*)x + rowc * (long long)(cols >> 2) + lane;
  const int iters = cols >> 7;  // 32 lanes x 4 floats = 128 floats per pass

  v4f a0 = {0.f, 0.f, 0.f, 0.f};
  v4f a1 = {0.f, 0.f, 0.f, 0.f};
  v4f a2 = {0.f, 0.f, 0.f, 0.f};
  v4f a3 = {0.f, 0.f, 0.f, 0.f};

#pragma unroll 2
  for (int i = 0; i < iters; i += 4) {
    v4f t0 = __builtin_nontemporal_load(src + (long long)(i + 0) * 32);
    v4f t1 = __builtin_nontemporal_load(src + (long long)(i + 1) * 32);
    v4f t2 = __builtin_nontemporal_load(src + (long long)(i + 2) * 32);
    v4f t3 = __builtin_nontemporal_load(src + (long long)(i + 3) * 32);
    a0 += t0;
    a1 += t1;
    a2 += t2;
    a3 += t3;
  }

  v4f s = (a0 + a1) + (a2 + a3);
  float p = (s.x + s.y) + (s.z + s.w);  // per-lane partial (128 elements)

  float total = wave_reduce_f32_wmma(p);

  if (lane == 0 && row < nrows) {
    // (1/sqrt(2))^12 == 2^-6
    out[row] = total * 0.015625f;
  }
}

extern "C" void kernel_launch(void* const* d_in, const int* in_sizes, int n_in,
                              void* d_out, int out_size, void* d_ws, size_t ws_size,
                              hipStream_t stream) {
  const float* x = (const float*)d_in[0];
  float* out = (float*)d_out;

  const int nrows = out_size;                 // 128*512 = 65536 rows
  const int cols  = in_sizes[0] / out_size;   // 4096

  const int rows_per_block = 8;               // 8 waves x 1 row each
  const int blocks = (nrows + rows_per_block - 1) / rows_per_block;

  WaveletActivation_rowsum_kernel<<<blocks, 256, 0, stream>>>(x, out, nrows, cols);
}